// Avg_Pooling_Variable_10806137717253
// MI455X (gfx1250) — compile-verified
//
#include <hip/hip_runtime.h>

// Prefix-mean pooling: out[b,d] = (1/eff_len[b]) * sum_{l < eff_len[b]} X[b,l,d]
// X: [128, 1024, 2048] fp32; lengths: [128] int32; out: [128, 2048] fp32.
//
// MI455X (gfx1250, wave32) strategy:
//  - Pure HBM-bound (1 GiB / 23.3 TB/s ~ 46 us full read). One wave owns one
//    (b, 16-wide d stripe); eff_len is wave-uniform and readfirstlane'd to an
//    SGPR, so all loop bounds are scalar (s_cmp/s_cbranch, EXEC stays all-1s
//    for WMMA) and rows >= eff_len are never read (~2x expected traffic cut).
//  - Accumulation on the matrix pipe: V_WMMA_F32_16X16X4_F32, A = mask
//    (ones / boundary mask), B = 4x16 fp32 feature tile. Two independent
//    accumulators break the WMMA->WMMA dependency chain; 8 NT loads per
//    unrolled iteration issue as one clause (1 KiB/wave/iter).
//  - C VGPR0 lanes 0..15 == out[d0..d0+15] -> contiguous 64B store per wave.

typedef float v2f __attribute__((ext_vector_type(2)));
typedef float v8f __attribute__((ext_vector_type(8)));

#define LL 1024
#define DD 2048

static __device__ __forceinline__ v8f wmma_acc(v2f a, v2f b, v8f c) {
    return __builtin_amdgcn_wmma_f32_16x16x4_f32(
        /*neg_a=*/false, a, /*neg_b=*/false, b,
        /*c_mod=*/(short)0, c, /*reuse_a=*/false, /*reuse_b=*/false);
}

__global__ __launch_bounds__(256) void prefix_mean_wmma(
    const float* __restrict__ features,
    const int* __restrict__ lengths,
    float* __restrict__ out)
{
    const int lane = threadIdx.x & 31;
    const int wave = blockIdx.x * (blockDim.x >> 5) + (threadIdx.x >> 5);

    // Wave-uniform -> force into SGPRs so loop control is scalar.
    const int b  = __builtin_amdgcn_readfirstlane(wave >> 7);         // 128 stripes / b
    const int d0 = __builtin_amdgcn_readfirstlane((wave & 127) << 4); // 16-wide stripe

    int len = lengths[b];                       // scalar load (uniform index)
    int eff = (len > 0) ? len : LL;             // eff in [1, 1024]
    eff = __builtin_amdgcn_readfirstlane(eff);

    // WMMA B-matrix (4x16 fp32, K x N) lane layout:
    //   lane i, reg j  <-  B[k = 2*(i>>4) + j, n = i & 15]
    const int n      = lane & 15;
    const int khalf  = lane >> 4;
    const int laneoff = 2 * khalf * DD + n;     // 32-bit per-lane element offset

    const float* __restrict__ bp =
        features + (size_t)b * (size_t)(LL * DD) + (size_t)d0;  // uniform base

    v8f c0 = {}, c1 = {};
    v2f ones; ones.x = 1.0f; ones.y = 1.0f;

    const int full4  = eff >> 2;    // complete K=4 chunks
    const int full16 = full4 >> 2;  // groups of 4 chunks (16 L-rows)

    int l = 0;
    for (int it = 0; it < full16; ++it, l += 16) {
        const float* q = bp + (size_t)l * DD + laneoff;
        v2f m0, m1, m2, m3;
        m0.x = __builtin_nontemporal_load(q + 0  * DD);
        m0.y = __builtin_nontemporal_load(q + 1  * DD);
        m1.x = __builtin_nontemporal_load(q + 4  * DD);
        m1.y = __builtin_nontemporal_load(q + 5  * DD);
        m2.x = __builtin_nontemporal_load(q + 8  * DD);
        m2.y = __builtin_nontemporal_load(q + 9  * DD);
        m3.x = __builtin_nontemporal_load(q + 12 * DD);
        m3.y = __builtin_nontemporal_load(q + 13 * DD);
        c0 = wmma_acc(ones, m0, c0);   // two independent accumulator chains
        c1 = wmma_acc(ones, m1, c1);
        c0 = wmma_acc(ones, m2, c0);
        c1 = wmma_acc(ones, m3, c1);
    }
    // Remaining complete 4-chunks (0..3 of them), scalar-bounded.
    for (int it = full16 << 2; it < full4; ++it, l += 4) {
        const float* q = bp + (size_t)l * DD + laneoff;
        v2f m;
        m.x = __builtin_nontemporal_load(q);
        m.y = __builtin_nontemporal_load(q + DD);
        c0 = wmma_acc(ones, m, c0);
    }
    // Boundary chunk: per-k mask in A. Rows l..l+3 are < L whenever rem != 0,
    // so the loads are in-bounds; masked rows contribute zero.
    const int rem = eff & 3;
    if (rem) {
        const float* q = bp + (size_t)l * DD + laneoff;
        v2f am;
        am.x = (2 * khalf + 0 < rem) ? 1.0f : 0.0f;
        am.y = (2 * khalf + 1 < rem) ? 1.0f : 0.0f;
        v2f m;
        m.x = __builtin_nontemporal_load(q);
        m.y = __builtin_nontemporal_load(q + DD);
        c0 = wmma_acc(am, m, c0);
    }

    // C layout: VGPR0, lanes 0..15 = C[M=0, N=lane] = sum for d = d0 + lane.
    const float r = (c0[0] + c1[0]) / (float)eff;
    if (lane < 16) {
        out[(size_t)b * DD + (size_t)(d0 + n)] = r;
    }
}

extern "C" void kernel_launch(void* const* d_in, const int* in_sizes, int n_in,
                              void* d_out, int out_size, void* d_ws, size_t ws_size,
                              hipStream_t stream) {
    const float* features = (const float*)d_in[0];
    const int*   lengths  = (const int*)d_in[1];
    float*       out      = (float*)d_out;

    // 16384 waves (128 b x 128 d-stripes) / 8 waves per 256-thread block.
    dim3 grid(2048), block(256);
    hipLaunchKernelGGL(prefix_mean_wmma, grid, block, 0, stream,
                       features, lengths, out);
}